// RPGGCN_43069932044897
// MI455X (gfx1250) — compile-verified
//
#include <hip/hip_runtime.h>
#include <math.h>

#define N_NODES 50000
#define N_EDGES 800000
#define F_IN    128
#define H_DIM   100
#define H_PAD   112     // 7 tiles of 16
#define C_DIM   16
#define LEAKY   0.01f

typedef float v2f __attribute__((ext_vector_type(2)));
typedef float v8f __attribute__((ext_vector_type(8)));

// ---------------------------------------------------------------- utilities
__global__ __launch_bounds__(256) void fill_kernel(float* __restrict__ p, float v, int n) {
    int i = blockIdx.x * blockDim.x + threadIdx.x;
    if (i < n) p[i] = v;
}

__global__ __launch_bounds__(256) void deg_kernel(const int* __restrict__ dst,
                                                  float* __restrict__ deg, int e) {
    int i = blockIdx.x * blockDim.x + threadIdx.x;
    if (i < e) atomicAdd(&deg[dst[i]], 1.0f);
}

__global__ __launch_bounds__(256) void dinv_kernel(const float* __restrict__ deg,
                                                   float* __restrict__ dinv, int n) {
    int i = blockIdx.x * blockDim.x + threadIdx.x;
    if (i < n) dinv[i] = rsqrtf(deg[i]);   // deg >= 1 always (self loop)
}

// ------------------------------------------------------- GEMM1: xw1 = X @ W1
// One wave computes one 16x16 output tile via V_WMMA_F32_16X16X4_F32 (full fp32).
// A (16x4 f32):  lanes 0-15 -> K=k0,k0+1 ; lanes 16-31 -> K=k0+2,k0+3
// B (4x16 f32):  VGPR r holds K=k0+2*half+r, N=n0+lane%16
// C (16x16 f32): VGPR r -> row m0 + r + 8*half, col n0 + lane%16
__global__ __launch_bounds__(256) void gemm1_wmma(const float* __restrict__ X,
                                                  const float* __restrict__ W1,
                                                  float* __restrict__ out /*[N,H_PAD]*/) {
    const int wave = (blockIdx.x * blockDim.x + threadIdx.x) >> 5;
    const int lane = threadIdx.x & 31;
    const int NT = H_PAD / 16;                       // 7
    const int TOT = (N_NODES / 16) * NT;             // 21875
    if (wave >= TOT) return;                         // wave-uniform: EXEC stays full
    const int m0 = (wave / NT) * 16;
    const int n0 = (wave % NT) * 16;
    const int half = lane >> 4, l16 = lane & 15;

    const float* aptr = X + (size_t)(m0 + l16) * F_IN + 2 * half;
    const int bcol = min(n0 + l16, H_DIM - 1);       // clamp padded cols (OOB safety)

    v8f acc = {};
#pragma unroll 4
    for (int k0 = 0; k0 < F_IN; k0 += 4) {
        v2f a = *(const v2f*)(aptr + k0);            // 8B-aligned
        v2f b;
        b.x = W1[(k0 + 2 * half + 0) * H_DIM + bcol];
        b.y = W1[(k0 + 2 * half + 1) * H_DIM + bcol];
        acc = __builtin_amdgcn_wmma_f32_16x16x4_f32(false, a, false, b,
                                                    (short)0, acc, false, false);
    }
    const int crow = m0 + 8 * half;
    const int ccol = n0 + l16;
#pragma unroll
    for (int r = 0; r < 8; ++r)
        out[(size_t)(crow + r) * H_PAD + ccol] = acc[r];
}

// ------------------------------------------------------- GEMM2: xw2 = h1 @ W2
// K = H_PAD (h1 padded cols are exact zeros), N = 16 exactly.
__global__ __launch_bounds__(256) void gemm2_wmma(const float* __restrict__ Hin,
                                                  const float* __restrict__ W2,
                                                  float* __restrict__ out /*[N,C]*/) {
    const int wave = (blockIdx.x * blockDim.x + threadIdx.x) >> 5;
    const int lane = threadIdx.x & 31;
    const int TOT = N_NODES / 16;                    // 3125
    if (wave >= TOT) return;
    const int m0 = wave * 16;
    const int half = lane >> 4, l16 = lane & 15;

    const float* aptr = Hin + (size_t)(m0 + l16) * H_PAD + 2 * half;

    v8f acc = {};
#pragma unroll 4
    for (int k0 = 0; k0 < H_PAD; k0 += 4) {
        v2f a = *(const v2f*)(aptr + k0);
        v2f b;
        int kr0 = min(k0 + 2 * half + 0, H_DIM - 1); // A is 0 for k>=100, so clamp is exact
        int kr1 = min(k0 + 2 * half + 1, H_DIM - 1);
        b.x = W2[kr0 * C_DIM + l16];
        b.y = W2[kr1 * C_DIM + l16];
        acc = __builtin_amdgcn_wmma_f32_16x16x4_f32(false, a, false, b,
                                                    (short)0, acc, false, false);
    }
    const int crow = m0 + 8 * half;
#pragma unroll
    for (int r = 0; r < 8; ++r)
        out[(size_t)(crow + r) * C_DIM + l16] = acc[r];
}

// -------------------------------------------- propagation 1: one wave per edge
__global__ __launch_bounds__(256) void prop1_kernel(const float* __restrict__ xw,
                                                    const int* __restrict__ src,
                                                    const int* __restrict__ dst,
                                                    const float* __restrict__ dinv,
                                                    float* __restrict__ h1) {
    const int wave = (blockIdx.x * blockDim.x + threadIdx.x) >> 5;
    const int lane = threadIdx.x & 31;
    if (wave >= N_EDGES) return;
    const int s = src[wave], d = dst[wave];
    const float nrm = dinv[s] * dinv[d];
    const float* xs = xw + (size_t)s * H_PAD;
    float* hd = h1 + (size_t)d * H_PAD;
    for (int c = lane; c < H_DIM; c += 32)           // L2-resident f32 atomics
        atomicAdd(&hd[c], xs[c] * nrm);
}

// self loop + bias + LeakyReLU; zero the padded columns for GEMM2
__global__ __launch_bounds__(256) void fin1_kernel(const float* __restrict__ xw,
                                                   const float* __restrict__ dinv,
                                                   const float* __restrict__ b1,
                                                   float* __restrict__ h1) {
    int i = blockIdx.x * blockDim.x + threadIdx.x;
    if (i >= N_NODES * H_PAD) return;
    const int node = i / H_PAD, c = i % H_PAD;
    float v = 0.0f;
    if (c < H_DIM) {
        const float di = dinv[node];
        v = h1[i] + xw[i] * (di * di) + b1[c];
        v = v > 0.0f ? v : LEAKY * v;
    }
    h1[i] = v;
}

// -------------------------------------------- propagation 2: thread per (edge, class)
__global__ __launch_bounds__(256) void prop2_kernel(const float* __restrict__ xw2,
                                                    const int* __restrict__ src,
                                                    const int* __restrict__ dst,
                                                    const float* __restrict__ dinv,
                                                    float* __restrict__ h2) {
    int i = blockIdx.x * blockDim.x + threadIdx.x;
    if (i >= N_EDGES * C_DIM) return;
    const int e = i >> 4, c = i & (C_DIM - 1);
    const int s = src[e], d = dst[e];
    const float nrm = dinv[s] * dinv[d];
    atomicAdd(&h2[(size_t)d * C_DIM + c], xw2[(size_t)s * C_DIM + c] * nrm);
}

// self loop + bias + log_softmax over the 16 classes, thread per node
__global__ __launch_bounds__(256) void fin2_kernel(const float* __restrict__ xw2,
                                                   const float* __restrict__ h2,
                                                   const float* __restrict__ dinv,
                                                   const float* __restrict__ b2,
                                                   float* __restrict__ out) {
    int i = blockIdx.x * blockDim.x + threadIdx.x;
    if (i >= N_NODES) return;
    const float di = dinv[i];
    const float sl = di * di;
    float v[C_DIM];
    float m = -INFINITY;
#pragma unroll
    for (int c = 0; c < C_DIM; ++c) {
        v[c] = h2[(size_t)i * C_DIM + c] + xw2[(size_t)i * C_DIM + c] * sl + b2[c];
        m = fmaxf(m, v[c]);
    }
    float sum = 0.0f;
#pragma unroll
    for (int c = 0; c < C_DIM; ++c) sum += __expf(v[c] - m);
    const float ls = __logf(sum) + m;
#pragma unroll
    for (int c = 0; c < C_DIM; ++c) out[(size_t)i * C_DIM + c] = v[c] - ls;
}

// ---------------------------------------------------------------- launcher
extern "C" void kernel_launch(void* const* d_in, const int* in_sizes, int n_in,
                              void* d_out, int out_size, void* d_ws, size_t ws_size,
                              hipStream_t stream) {
    const float* x   = (const float*)d_in[0];
    const float* W1  = (const float*)d_in[1];
    const float* b1  = (const float*)d_in[2];
    const float* W2  = (const float*)d_in[3];
    const float* b2  = (const float*)d_in[4];
    const int*   ei  = (const int*)d_in[5];
    const int*   src = ei;
    const int*   dst = ei + N_EDGES;
    float* out = (float*)d_out;

    // workspace layout (floats)
    float* ws   = (float*)d_ws;
    float* xw1  = ws;                                 // [N, H_PAD]   5.6M
    float* h1   = xw1 + (size_t)N_NODES * H_PAD;      // [N, H_PAD]   5.6M
    float* xw2  = h1  + (size_t)N_NODES * H_PAD;      // [N, C]       0.8M
    float* h2   = xw2 + (size_t)N_NODES * C_DIM;      // [N, C]       0.8M
    float* deg  = h2  + (size_t)N_NODES * C_DIM;      // [N]
    float* dinv = deg + N_NODES;                      // [N]

    const int B = 256;
    // init accumulators / degrees (every call: no cross-call state)
    fill_kernel<<<(N_NODES * H_PAD + B - 1) / B, B, 0, stream>>>(h1, 0.0f, N_NODES * H_PAD);
    fill_kernel<<<(N_NODES * C_DIM + B - 1) / B, B, 0, stream>>>(h2, 0.0f, N_NODES * C_DIM);
    fill_kernel<<<(N_NODES + B - 1) / B, B, 0, stream>>>(deg, 1.0f, N_NODES); // self loop
    deg_kernel<<<(N_EDGES + B - 1) / B, B, 0, stream>>>(dst, deg, N_EDGES);
    dinv_kernel<<<(N_NODES + B - 1) / B, B, 0, stream>>>(deg, dinv, N_NODES);

    // layer 1
    const int tiles1 = (N_NODES / 16) * (H_PAD / 16);           // 21875 waves
    gemm1_wmma<<<(tiles1 + 7) / 8, B, 0, stream>>>(x, W1, xw1);
    prop1_kernel<<<N_EDGES / 8, B, 0, stream>>>(xw1, src, dst, dinv, h1);
    fin1_kernel<<<(N_NODES * H_PAD + B - 1) / B, B, 0, stream>>>(xw1, dinv, b1, h1);

    // layer 2
    const int tiles2 = N_NODES / 16;                            // 3125 waves
    gemm2_wmma<<<(tiles2 + 7) / 8, B, 0, stream>>>(h1, W2, xw2);
    prop2_kernel<<<(N_EDGES * C_DIM + B - 1) / B, B, 0, stream>>>(xw2, src, dst, dinv, h2);
    fin2_kernel<<<(N_NODES + B - 1) / B, B, 0, stream>>>(xw2, h2, dinv, b2, out);
}